// SegLoss_pathch_51917564674638
// MI455X (gfx1250) — compile-verified
//
#include <hip/hip_runtime.h>

typedef __attribute__((ext_vector_type(2))) float v2f;
typedef __attribute__((ext_vector_type(8))) float v8f;

#define DVOL 256
#define PATCHES_PER_WAVE 16
#define WAVES_PER_BLOCK 8

// Compute c_t value for neighborhood element e (0..26) of a patch centered at
// flat index `base`. e -> (d0,d1,d2) in {-1,0,1}^3, row-major [D,D,D].
__device__ __forceinline__ float patch_val(const float* __restrict__ a,
                                           const float* __restrict__ b,
                                           const float* __restrict__ ts,
                                           int base, int e,
                                           float sa, float mina,
                                           float sb, float minb) {
    const int q9 = e / 9;
    const int r9 = e - q9 * 9;
    const int q3 = r9 / 3;
    const int d0 = q9 - 1;
    const int d1 = q3 - 1;
    const int d2 = (r9 - q3 * 3) - 1;
    const int f  = base + d0 * (DVOL * DVOL) + d1 * DVOL + d2;
    const float av = a[f] * sa + mina;
    const float bv = b[f] * sb + minb;
    return (av / bv) * ts[f];
}

// One wave32 handles 16 patches. Rows of the A matrix (16x4 f32) are patches;
// B is all-ones so every column of D = rowsum(A) regardless of B layout.
// Three f32 WMMA accumulators carry S1=sum(c), S2=sum(c^2), N=count(c!=0)
// across 7 K=4 chunks (27 elements + 1 zero pad) at full f32 precision.
__global__ void __launch_bounds__(256)
seg_cor_kernel(const float* __restrict__ a, const float* __restrict__ b,
               const float* __restrict__ ts,
               const float* __restrict__ mina_p, const float* __restrict__ maxa_p,
               const float* __restrict__ minb_p, const float* __restrict__ maxb_p,
               const int* __restrict__ pidx,
               float* __restrict__ partials, int nsample) {
    const int lane = (int)(threadIdx.x & 31u);
    const int wave = (int)((blockIdx.x * blockDim.x + threadIdx.x) >> 5);
    const int m    = lane & 15;   // A-matrix row (patch within wave)
    const int h    = lane >> 4;   // lane half selects K pair {2h, 2h+1}
    const int p    = wave * PATCHES_PER_WAVE + m;
    const bool valid = (p < nsample);

    const float mina = *mina_p;
    const float sa   = *maxa_p - mina;
    const float minb = *minb_p;
    const float sb   = *maxb_p - minb;

    int base = 0;
    if (valid) {
        const int i0 = pidx[3 * p + 0];
        const int i1 = pidx[3 * p + 1];
        const int i2 = pidx[3 * p + 2];
        base = (i0 * DVOL + i1) * DVOL + i2;
    }

    v2f bones; bones.x = 1.0f; bones.y = 1.0f;   // B = all ones (layout-proof)
    v8f accS1 = {}; v8f accS2 = {}; v8f accN = {};

    #pragma unroll
    for (int t = 0; t < 7; ++t) {
        const int e0 = t * 4 + h * 2;            // this lane's K elements
        float c0 = 0.0f, c1 = 0.0f;
        if (valid) {
            c0 = patch_val(a, b, ts, base, e0, sa, mina, sb, minb);
            if (e0 + 1 < 27)                     // only chunk 6, half 1 pads
                c1 = patch_val(a, b, ts, base, e0 + 1, sa, mina, sb, minb);
        }
        v2f A1; A1.x = c0;                      A1.y = c1;
        v2f A2; A2.x = c0 * c0;                 A2.y = c1 * c1;
        v2f AM; AM.x = (c0 != 0.0f) ? 1.0f : 0.0f;
                AM.y = (c1 != 0.0f) ? 1.0f : 0.0f;
        // 8-arg pattern: (neg_a, A, neg_b, B, c_mod, C, reuse_a, reuse_b)
        accS1 = __builtin_amdgcn_wmma_f32_16x16x4_f32(false, A1, false, bones,
                                                      (short)0, accS1, false, false);
        accS2 = __builtin_amdgcn_wmma_f32_16x16x4_f32(false, A2, false, bones,
                                                      (short)0, accS2, false, false);
        accN  = __builtin_amdgcn_wmma_f32_16x16x4_f32(false, AM, false, bones,
                                                      (short)0, accN,  false, false);
    }

    // C/D layout: acc element v in lane l holds D[v + 8*(l>=16)][l&15]; every
    // column is identical, so each lane sees 8 of the 16 patch sums.
    float part = 0.0f;
    #pragma unroll
    for (int v = 0; v < 8; ++v) {
        const int M  = v + 8 * h;
        const int pp = wave * PATCHES_PER_WAVE + M;
        if (pp < nsample) {
            const float S1 = accS1[v];
            const float S2 = accS2[v];
            const float n  = accN[v];
            const float mean = S1 / n;
            const float var  = (S2 - S1 * mean) / (n - 1.0f); // = (S2 - S1^2/n)/(n-1)
            part += var / mean;
        }
    }
    // Merge the two lane halves (M=0..7 plus M=8..15) -> wave partial.
    part += __shfl_xor(part, 16, 32);
    if (lane == 0) partials[wave] = part;
}

// Deterministic single-wave tree reduction of per-wave partials.
__global__ void seg_reduce_kernel(const float* __restrict__ partials, int n,
                                  float* __restrict__ out, float inv_nsample) {
    const int lane = (int)(threadIdx.x & 31u);
    float s = 0.0f;
    for (int i = lane; i < n; i += 32) s += partials[i];
    #pragma unroll
    for (int off = 16; off > 0; off >>= 1) s += __shfl_xor(s, off, 32);
    if (lane == 0) *out = s * inv_nsample;
}

extern "C" void kernel_launch(void* const* d_in, const int* in_sizes, int n_in,
                              void* d_out, int out_size, void* d_ws, size_t ws_size,
                              hipStream_t stream) {
    const float* a    = (const float*)d_in[0];
    const float* b    = (const float*)d_in[1];
    const float* ts   = (const float*)d_in[2];
    const float* mina = (const float*)d_in[3];
    const float* maxa = (const float*)d_in[4];
    const float* minb = (const float*)d_in[5];
    const float* maxb = (const float*)d_in[6];
    const int*   pidx = (const int*)d_in[7];

    const int nsample = in_sizes[7] / 3;                               // 5000
    const int nwaves  = (nsample + PATCHES_PER_WAVE - 1) / PATCHES_PER_WAVE; // 313
    const int nblocks = (nwaves + WAVES_PER_BLOCK - 1) / WAVES_PER_BLOCK;    // 40
    const int nparts  = nblocks * WAVES_PER_BLOCK;                     // 320

    float* partials = (float*)d_ws;   // 320 floats; fully rewritten every call

    seg_cor_kernel<<<nblocks, 256, 0, stream>>>(a, b, ts, mina, maxa, minb, maxb,
                                                pidx, partials, nsample);
    seg_reduce_kernel<<<1, 32, 0, stream>>>(partials, nparts, (float*)d_out,
                                            1.0f / (float)nsample);
}